// MoluculaCNN_27324581937482
// MI455X (gfx1250) — compile-verified
//
#include <hip/hip_runtime.h>
#include <math.h>

typedef __attribute__((ext_vector_type(2))) float v2f;
typedef __attribute__((ext_vector_type(8))) float v8f;

#define TPB 256

// ---------------- degree / normalization precompute ----------------

__global__ void init_deg_kernel(float* degW, float* deg1, int N) {
  int i = blockIdx.x * blockDim.x + threadIdx.x;
  if (i < N) { degW[i] = 1.0f; deg1[i] = 1.0f; }   // self-loop weight 1
}

__global__ void acc_deg_kernel(const int* __restrict__ dst, const float* __restrict__ ew,
                               float* degW, float* deg1, int E) {
  int e = blockIdx.x * blockDim.x + threadIdx.x;
  if (e < E) {
    int d = dst[e];
    atomicAdd(&degW[d], ew[e]);
    atomicAdd(&deg1[d], 1.0f);
  }
}

__global__ void rsqrt_deg_kernel(float* degW, float* deg1, int N) {
  int i = blockIdx.x * blockDim.x + threadIdx.x;
  if (i < N) { degW[i] = rsqrtf(degW[i]); deg1[i] = rsqrtf(deg1[i]); }
}

__global__ void norm_kernel(const int* __restrict__ src, const int* __restrict__ dst,
                            const float* __restrict__ ew,
                            const float* __restrict__ dinvW, const float* __restrict__ dinv1,
                            float* __restrict__ normW, float* __restrict__ norm1, int E) {
  int e = blockIdx.x * blockDim.x + threadIdx.x;
  if (e < E) {
    int s = src[e], d = dst[e];
    normW[e] = dinvW[s] * ew[e] * dinvW[d];
    norm1[e] = dinv1[s] * dinv1[d];
  }
}

// ---------------- dense GEMM via fp32 WMMA (16x16x4) ----------------
// One wave32 per 16x16 output tile.  A: [N,K] row-major, W: [K,C] row-major.
// A 16x4 f32 operand layout (ISA 7.12.2): lanes 0-15 hold row M=lane, K=k0..k0+1;
// lanes 16-31 hold row M=lane-16, K=k0+2..k0+3.  B 4x16 mirrors with rows across
// lanes.  C/D 16x16 f32: VGPR r = rows M=r (lanes 0-15) and M=r+8 (lanes 16-31).
//
// Row guard is a CLAMP, not predication: A-row in lane l only feeds output row
// m0+l, which is simply not stored when out of range.  EXEC stays all-ones
// (required by WMMA); inner loop is branch-free: 1x b64 + 2x b32 loads + 1 wmma
// per K-step.  Store epilogue takes an unconditional fast path for full tiles
// (always, since N % 16 == 0 here) to avoid per-store exec masking.

__global__ __launch_bounds__(32) void gemm_wmma_kernel(const float* __restrict__ A,
                                                       const float* __restrict__ W,
                                                       float* __restrict__ out,
                                                       int N, int K, int C) {
  int tilesN = C >> 4;
  int m0 = (blockIdx.x / tilesN) << 4;
  int n0 = (blockIdx.x % tilesN) << 4;
  int lane = threadIdx.x & 31;
  int half = lane >> 4;       // 0: K pair {0,1}, rows M=r ; 1: K pair {2,3}, rows M=r+8
  int l    = lane & 15;

  int rowA = m0 + l;
  if (rowA >= N) rowA = N - 1;          // clamp (see comment above)

  const float* pa  = A + (size_t)rowA * K + (half << 1);       // 8B aligned (K % 4 == 0)
  const float* pb0 = W + (size_t)(half << 1) * C + n0 + l;
  const float* pb1 = pb0 + C;
  const int    sb  = C << 2;                                   // 4*C floats per K-step

  v8f c = {};
#pragma unroll 2
  for (int k0 = 0; k0 < K; k0 += 4) {
    v2f a = *(const v2f*)pa;
    v2f b;
    b.x = *pb0;
    b.y = *pb1;
    c = __builtin_amdgcn_wmma_f32_16x16x4_f32(false, a, false, b, (short)0, c, false, false);
    pa  += 4;
    pb0 += sb;
    pb1 += sb;
  }

  int rbase = m0 + (half << 3);
  float* po = out + (size_t)rbase * C + n0 + l;
  if (rbase + 8 <= N) {
    // full tile: unconditional coalesced stores, no exec-mask toggling
#pragma unroll
    for (int r = 0; r < 8; ++r) po[(size_t)r * C] = c[r];
  } else {
#pragma unroll
    for (int r = 0; r < 8; ++r) {
      if (rbase + r < N) po[(size_t)r * C] = c[r];
    }
  }
}

// ---------------- propagation: self term + edge scatter ----------------

__global__ void prop_self_kernel(const float* __restrict__ xw, const float* __restrict__ dinv,
                                 const float* __restrict__ b, float* __restrict__ out,
                                 int total, int lc) {
  int idx = blockIdx.x * blockDim.x + threadIdx.x;
  if (idx < total) {
    int i = idx >> lc;
    int c = idx & ((1 << lc) - 1);
    float di = dinv[i];
    out[idx] = di * di * xw[idx] + b[c];
  }
}

__global__ void prop_scatter_kernel(const float* __restrict__ xw, const int* __restrict__ src,
                                    const int* __restrict__ dst, const float* __restrict__ norm,
                                    float* out, int E, int lc) {
  int t = blockIdx.x * blockDim.x + threadIdx.x;
  int e = t >> lc;
  if (e < E) {
    int c = t & ((1 << lc) - 1);
    atomicAdd(&out[((long)dst[e] << lc) + c], norm[e] * xw[((long)src[e] << lc) + c]);
  }
}

// ---------------- column sum (LDS tree reduction, no atomics) ----------------

__global__ void colsum_kernel(const float* __restrict__ x, float* __restrict__ cs,
                              int N, int C) {
  __shared__ float sm[TPB];
  int c = blockIdx.x;  // one block per channel
  float s = 0.0f;
  for (int i = threadIdx.x; i < N; i += blockDim.x) s += x[(long)i * C + c];
  sm[threadIdx.x] = s;
  __syncthreads();
  for (int off = TPB / 2; off > 0; off >>= 1) {
    if (threadIdx.x < off) sm[threadIdx.x] += sm[threadIdx.x + off];
    __syncthreads();
  }
  if (threadIdx.x == 0) cs[c] = sm[0];
}

// ---------------- fused broadcast-sum + ELU epilogue ----------------
// s2 = xc + colsum(xb) + N * colsum(xa);  out = elu(s2)

__global__ void final_elu_kernel(const float* __restrict__ xc,
                                 const float* __restrict__ csA, const float* __restrict__ csB,
                                 float* __restrict__ out, int total, float Nf) {
  int idx = blockIdx.x * blockDim.x + threadIdx.x;
  if (idx < total) {
    int c = idx & 31;
    float v = xc[idx] + csB[c] + Nf * csA[c];
    out[idx] = (v > 0.0f) ? v : (expf(v) - 1.0f);
  }
}

// ---------------- driver ----------------

extern "C" void kernel_launch(void* const* d_in, const int* in_sizes, int n_in,
                              void* d_out, int out_size, void* d_ws, size_t ws_size,
                              hipStream_t stream) {
  const float* x  = (const float*)d_in[0];
  const int*   src = (const int*)d_in[1];
  const int*   dst = (const int*)d_in[2];
  const float* ew = (const float*)d_in[3];
  const float* W1 = (const float*)d_in[4];  const float* b1 = (const float*)d_in[5];
  const float* W2 = (const float*)d_in[6];  const float* b2 = (const float*)d_in[7];
  const float* W3 = (const float*)d_in[8];  const float* b3 = (const float*)d_in[9];
  const float* W4 = (const float*)d_in[10]; const float* b4 = (const float*)d_in[11];

  const int N = in_sizes[0] / 8;   // C_IN = 8
  const int E = in_sizes[1];

  float* ws = (float*)d_ws;
  size_t off = 0;
  float* dinvW = ws + off; off += N;            // degW -> dinvW in place
  float* dinv1 = ws + off; off += N;
  float* normW = ws + off; off += E;
  float* norm1 = ws + off; off += E;
  float* B0 = ws + off; off += (size_t)N * 256; // gemm output (xw)
  float* B1 = ws + off; off += (size_t)N * 256; // wide features
  float* B2 = ws + off; off += (size_t)N * 256; // narrow features / xa,xb,xc
  float* csA = ws + off; off += 32;
  float* csB = ws + off; off += 32;

  // --- normalization coefficients, computed once (reference recomputes 7x) ---
  init_deg_kernel<<<(N + TPB - 1) / TPB, TPB, 0, stream>>>(dinvW, dinv1, N);
  acc_deg_kernel<<<(E + TPB - 1) / TPB, TPB, 0, stream>>>(dst, ew, dinvW, dinv1, E);
  rsqrt_deg_kernel<<<(N + TPB - 1) / TPB, TPB, 0, stream>>>(dinvW, dinv1, N);
  norm_kernel<<<(E + TPB - 1) / TPB, TPB, 0, stream>>>(src, dst, ew, dinvW, dinv1,
                                                       normW, norm1, E);

  auto layer = [&](const float* xin, const float* W, const float* b,
                   const float* norm, const float* dinv,
                   float* outF, int K, int C, int lc) {
    int tilesM = (N + 15) / 16;
    gemm_wmma_kernel<<<tilesM * (C >> 4), 32, 0, stream>>>(xin, W, B0, N, K, C);
    int tot = N * C;
    prop_self_kernel<<<(tot + TPB - 1) / TPB, TPB, 0, stream>>>(B0, dinv, b, outF, tot, lc);
    int etot = E * C;  // max 320000*256 = 81.9M < 2^31
    prop_scatter_kernel<<<(etot + TPB - 1) / TPB, TPB, 0, stream>>>(B0, src, dst, norm,
                                                                    outF, E, lc);
  };

  // conv1: 8 -> 128 (edge-weighted), conv2: 128 -> 32
  layer(x,  W1, b1, normW, dinvW, B1,   8, 128, 7);
  layer(B1, W2, b2, norm1, dinv1, B2, 128,  32, 5);

  // CNN2: 32 -> 256 -> 32  => xa (in B2), take its column sum immediately
  layer(B2, W3, b3, norm1, dinv1, B1,  32, 256, 8);
  layer(B1, W4, b4, norm1, dinv1, B2, 256,  32, 5);
  colsum_kernel<<<32, TPB, 0, stream>>>(B2, csA, N, 32);

  // CNN3 => xb
  layer(B2, W3, b3, norm1, dinv1, B1,  32, 256, 8);
  layer(B1, W4, b4, norm1, dinv1, B2, 256,  32, 5);
  colsum_kernel<<<32, TPB, 0, stream>>>(B2, csB, N, 32);

  // CNN4 => xc
  layer(B2, W3, b3, norm1, dinv1, B1,  32, 256, 8);
  layer(B1, W4, b4, norm1, dinv1, B2, 256,  32, 5);

  // out = elu(xc + colsum(xb) + N*colsum(xa))
  final_elu_kernel<<<(N * 32 + TPB - 1) / TPB, TPB, 0, stream>>>(B2, csA, csB,
                                                                 (float*)d_out, N * 32,
                                                                 (float)N);
}